// DotProductSparseAttention_66675072303640
// MI455X (gfx1250) — compile-verified
//
#include <hip/hip_runtime.h>

typedef float v2f __attribute__((ext_vector_type(2)));
typedef float v8f __attribute__((ext_vector_type(8)));

#define B_    4
#define H_    16
#define S_    2048
#define D_    64
#define QT    16            // queries per block
#define WAVES 8
#define KSLICE (S_ / WAVES) // 256 keys per wave

__launch_bounds__(256)
__global__ void topk_attn_kernel(const float* __restrict__ Q,
                                 const float* __restrict__ K,
                                 const float* __restrict__ V,
                                 const int*   __restrict__ numk,
                                 float* __restrict__ O) {
  __shared__ float sc[QT * S_];    // 128 KB: scores -> probabilities (in place)
  __shared__ float obuf[QT * D_];  // 4 KB: cross-wave output accumulation

  const int tid  = threadIdx.x;
  const int lane = tid & 31;
  const int wave = tid >> 5;
  const int m    = lane & 15;   // M (A/C row) and N (B/C col) index for this lane
  const int hi   = lane >> 4;   // which half of the wave (K-split in A/B frags)

  const int qtile = blockIdx.x & 127;   // S_/QT = 128 query tiles
  const int bh    = blockIdx.x >> 7;    // 0..63 (b*H + h)
  const int q0    = qtile * QT;
  const long base = (long)bh * S_ * D_;

  // ------------- Phase 1: scores = (Q K^T) / sqrt(D) -------------
  // Preload this lane's Q fragments for all 16 d-steps (row m, halves per hi).
  float qa[16][2];
#pragma unroll
  for (int i = 0; i < 16; ++i) {
    const int d0 = i * 4 + hi * 2;
    const float* qp = Q + base + (long)(q0 + m) * D_ + d0;
    qa[i][0] = qp[0];
    qa[i][1] = qp[1];
  }

  const int kw = wave * KSLICE;
#pragma unroll 1
  for (int t = 0; t < KSLICE / 16; ++t) {
    const int kb = kw + t * 16;
    v8f acc = {0.f, 0.f, 0.f, 0.f, 0.f, 0.f, 0.f, 0.f};
    const float* kp = K + base + (long)(kb + m) * D_;   // n = m for B frag
#pragma unroll
    for (int i = 0; i < 16; ++i) {
      const int d0 = i * 4 + hi * 2;
      v2f a; a.x = qa[i][0]; a.y = qa[i][1];
      v2f b; b.x = kp[d0];   b.y = kp[d0 + 1];
      acc = __builtin_amdgcn_wmma_f32_16x16x4_f32(false, a, false, b,
                                                  (short)0, acc, false, false);
    }
#pragma unroll
    for (int r = 0; r < 8; ++r) {
      const int row = r + hi * 8;
      sc[row * S_ + kb + m] = acc[r] * 0.125f;  // 1/sqrt(64)
    }
  }
  __syncthreads();

  // ------------- Phase 2: per-row top-k threshold + softmax -------------
  int nk = numk[0];
  if (nk > S_) nk = S_;

#pragma unroll 1
  for (int rr = wave * 2; rr < wave * 2 + 2; ++rr) {
    float cache[64];                       // 2048 row elems = 32 lanes x 64
#pragma unroll
    for (int j = 0; j < 64; ++j)
      cache[j] = sc[rr * S_ + j * 32 + lane];

    float mx = -3.0e38f, mn = 3.0e38f;
#pragma unroll
    for (int j = 0; j < 64; ++j) { mx = fmaxf(mx, cache[j]); mn = fminf(mn, cache[j]); }
#pragma unroll
    for (int off = 16; off > 0; off >>= 1) {
      mx = fmaxf(mx, __shfl_xor(mx, off, 32));
      mn = fminf(mn, __shfl_xor(mn, off, 32));
    }

    // Bisection for the nk-th largest value T: invariant count(>=lo) >= nk > count(>=hs)
    float lo = mn;
    if (nk > 0) {
      float hs = mx + 1.0e-3f + fabsf(mx) * 1.0e-6f;
#pragma unroll 1
      for (int it = 0; it < 32; ++it) {
        const float mid = 0.5f * (lo + hs);
        int cnt = 0;
#pragma unroll
        for (int j = 0; j < 64; ++j) cnt += (cache[j] >= mid) ? 1 : 0;
#pragma unroll
        for (int off = 16; off > 0; off >>= 1) cnt += __shfl_xor(cnt, off, 32);
        if (cnt >= nk) lo = mid; else hs = mid;
      }
    } else {
      lo = -3.0e38f;  // nk==0: reference skips masking
    }

    float sum = 0.f;
#pragma unroll
    for (int j = 0; j < 64; ++j) {
      const float x = cache[j];
      const float e = (x >= lo) ? __expf(x - mx) : 0.0f;
      cache[j] = e;
      sum += e;
    }
#pragma unroll
    for (int off = 16; off > 0; off >>= 1) sum += __shfl_xor(sum, off, 32);
    const float inv = 1.0f / sum;
#pragma unroll
    for (int j = 0; j < 64; ++j)
      sc[rr * S_ + j * 32 + lane] = cache[j] * inv;
  }

  // zero the cross-wave output accumulator
  for (int i = tid; i < QT * D_; i += 256) obuf[i] = 0.f;
  __syncthreads();

  // ------------- Phase 3: O = P V -------------
#pragma unroll 1
  for (int nt = 0; nt < 4; ++nt) {
    v8f acc = {0.f, 0.f, 0.f, 0.f, 0.f, 0.f, 0.f, 0.f};
#pragma unroll 4
    for (int s = 0; s < KSLICE / 4; ++s) {
      const int k0 = kw + s * 4 + hi * 2;
      v2f a; a.x = sc[m * S_ + k0]; a.y = sc[m * S_ + k0 + 1];
      const float* vp = V + base + (long)k0 * D_ + nt * 16 + m;  // n = m
      v2f b; b.x = vp[0]; b.y = vp[D_];
      acc = __builtin_amdgcn_wmma_f32_16x16x4_f32(false, a, false, b,
                                                  (short)0, acc, false, false);
    }
#pragma unroll
    for (int r = 0; r < 8; ++r) {
      const int row = r + hi * 8;
      atomicAdd(&obuf[row * D_ + nt * 16 + m], acc[r]);   // ds_add_f32
    }
  }
  __syncthreads();

  // coalesced store of the 16x64 output tile
  for (int i = tid; i < QT * D_; i += 256) {
    const int row = i / D_;
    const int col = i % D_;
    O[base + (long)(q0 + row) * D_ + col] = obuf[i];
  }
}

extern "C" void kernel_launch(void* const* d_in, const int* in_sizes, int n_in,
                              void* d_out, int out_size, void* d_ws, size_t ws_size,
                              hipStream_t stream) {
  (void)in_sizes; (void)n_in; (void)d_ws; (void)ws_size; (void)out_size;
  const float* Q   = (const float*)d_in[0];
  const float* K   = (const float*)d_in[1];
  const float* V   = (const float*)d_in[2];
  const int*   nk  = (const int*)d_in[3];
  float*       O   = (float*)d_out;

  const int blocks = B_ * H_ * (S_ / QT);  // 8192
  topk_attn_kernel<<<dim3(blocks), dim3(256), 0, stream>>>(Q, K, V, nk, O);
}